// W2LossConst_81690277970115
// MI455X (gfx1250) — compile-verified
//
#include <hip/hip_runtime.h>
#include <hip/hip_bf16.h>

// Problem constants (match the reference: B=2048, NT=8192, K=128).
constexpr int kNT     = 8192;
constexpr int kK      = 128;
constexpr int TILE    = 1024;            // f elements staged per LDS tile
constexpr int NTILES  = kNT / TILE;      // 8
constexpr int THREADS = 256;             // 8 wave32s
constexpr int NWAVES  = THREADS / 32;

// ---------------------------------------------------------------------------
// CDNA5 async global->LDS copy (GVS mode: SGPR64 base + per-lane 32-bit byte
// offset) and the split asynccnt waits. Inline asm so it works on both
// toolchains regardless of builtin arity.
// ---------------------------------------------------------------------------
__device__ __forceinline__ void async_tile_b128(const float* gbase,
                                                unsigned gbyte_off,
                                                unsigned lds_byte_addr) {
#if defined(__HIP_DEVICE_COMPILE__) && defined(__gfx1250__)
    asm volatile("global_load_async_to_lds_b128 %0, %1, %2 offset:0"
                 :
                 : "v"(lds_byte_addr), "v"(gbyte_off), "s"(gbase)
                 : "memory");
#else
    (void)gbase; (void)gbyte_off; (void)lds_byte_addr;
#endif
}

__device__ __forceinline__ void wait_async_le1() {
#if defined(__HIP_DEVICE_COMPILE__) && defined(__gfx1250__)
    asm volatile("s_wait_asynccnt 0x1" ::: "memory");
#endif
}
__device__ __forceinline__ void wait_async_le0() {
#if defined(__HIP_DEVICE_COMPILE__) && defined(__gfx1250__)
    asm volatile("s_wait_asynccnt 0x0" ::: "memory");
#endif
}

__device__ __forceinline__ unsigned lds_addr_of(const void* p) {
    // addrspacecast(LDS -> generic) puts the LDS byte offset in the low 32 bits
    return (unsigned)(uintptr_t)p;
}

// ---------------------------------------------------------------------------
// One block per trace. Streams f[b,:] through double-buffered async LDS tiles,
// does a hierarchical prefix scan (thread(4) -> wave32 shuffle -> block LDS),
// evaluates the monotone cubic spline from LDS-resident tables, accumulates
// the weighted trapezoid sum.
// ---------------------------------------------------------------------------
__global__ __launch_bounds__(THREADS, 1) void w2loss_kernel(
    const float* __restrict__ f,       // [B, NT]
    const float* __restrict__ t,       // [NT]
    const float* __restrict__ knots,   // [B, K]
    const float* __restrict__ coeffs,  // [B, K-1, 4]
    float* __restrict__ out)           // [B]
{
    __shared__ __align__(16) float s_tile[2][TILE];
    __shared__ float s_guard[2];               // f value preceding each tile
    __shared__ float s_knots[kK];
    __shared__ __align__(16) float s_coef[(kK - 1) * 4];
    __shared__ float s_wsum[NWAVES];
    __shared__ float s_red[NWAVES];

    const int b    = blockIdx.x;
    const int tid  = threadIdx.x;
    const int lane = tid & 31;
    const int wv   = tid >> 5;

    const float* ftr = f + (size_t)b * kNT;
    const float  dt  = t[1] - t[0];

    // ---- Prefetch tiles 0 and 1 (async, 16B per lane, 4KB per tile). -------
    {
        const unsigned lbyte = (unsigned)(tid * 16);
        async_tile_b128(ftr, lbyte,              lds_addr_of(&s_tile[0][0]) + lbyte);
        async_tile_b128(ftr, TILE * 4u + lbyte,  lds_addr_of(&s_tile[1][0]) + lbyte);
    }

    // ---- Stage per-trace spline tables (small; regular path). --------------
    if (tid < kK) s_knots[tid] = knots[(size_t)b * kK + tid];
    for (int i = tid; i < (kK - 1) * 4; i += THREADS)
        s_coef[i] = coeffs[(size_t)b * (kK - 1) * 4 + i];
    if (tid == 0) s_guard[0] = 0.0f;   // j==0 increment is forced to 0 anyway

    float carry = 0.0f;   // CDF value entering the current tile
    float acc   = 0.0f;   // per-thread partial of sum_j w_j*(t_j - S(F_j))^2*f_j

    for (int k = 0; k < NTILES; ++k) {
        const int cbuf = k & 1;

        // Tile k's async copy done (in-order per wave), then make it WGP-visible.
        if (k + 1 < NTILES) wait_async_le1(); else wait_async_le0();
        __syncthreads();                                       // (A)

        if (tid == 0) s_guard[cbuf ^ 1] = s_tile[cbuf][TILE - 1];

        // ---- trapezoid increments + thread-local inclusive scan (4 elems) --
        const int    j0  = k * TILE + 4 * tid;                 // global index
        const float4 fv  = *(const float4*)&s_tile[cbuf][4 * tid];
        const float  fm1 = (tid == 0) ? s_guard[cbuf] : s_tile[cbuf][4 * tid - 1];
        const float i0 = (j0 == 0) ? 0.0f : 0.5f * (fv.x + fm1) * dt;
        const float i1 = 0.5f * (fv.y + fv.x) * dt;
        const float i2 = 0.5f * (fv.z + fv.y) * dt;
        const float i3 = 0.5f * (fv.w + fv.z) * dt;
        const float p0 = i0, p1 = p0 + i1, p2 = p1 + i2, p3 = p2 + i3;

        // ---- wave32 inclusive shuffle scan of thread totals ----------------
        float s = p3;
        #pragma unroll
        for (int off = 1; off < 32; off <<= 1) {
            const float n = __shfl_up(s, off, 32);
            if (lane >= off) s += n;
        }
        if (lane == 31) s_wsum[wv] = s;
        const float excl = s - p3;            // earlier lanes in this wave
        __syncthreads();                                       // (B)

        float wpre = 0.0f, tot = 0.0f;
        #pragma unroll
        for (int w = 0; w < NWAVES; ++w) {
            const float v = s_wsum[w];
            if (w < wv) wpre += v;
            tot += v;
        }
        const float base = carry + wpre + excl;
        carry += tot;

        const float F0 = base + p0, F1 = base + p1, F2 = base + p2, F3 = base + p3;

        // ---- spline interval: binary search for F0, monotone walks after ---
        int idx = 0;
        #pragma unroll
        for (int st = 64; st; st >>= 1) {
            const int c = idx + st;            // c <= 127 always
            if (s_knots[c] <= F0) idx = c;
        }
        if (idx > kK - 2) idx = kK - 2;
        float kv = s_knots[idx];
        const int   id0 = idx;  const float kv0 = kv;
        auto walk = [&](float q) {
            while (idx < kK - 2) {
                const float x = s_knots[idx + 1];
                if (x <= q) { ++idx; kv = x; } else break;
            }
        };
        walk(F1); const int id1 = idx; const float kv1 = kv;
        walk(F2); const int id2 = idx; const float kv2 = kv;
        walk(F3); const int id3 = idx; const float kv3 = kv;

        const float4 c0 = *(const float4*)&s_coef[4 * id0];
        const float4 c1 = (id1 == id0) ? c0 : *(const float4*)&s_coef[4 * id1];
        const float4 c2 = (id2 == id1) ? c1 : *(const float4*)&s_coef[4 * id2];
        const float4 c3 = (id3 == id2) ? c2 : *(const float4*)&s_coef[4 * id3];

        auto spl = [](float q, float kvv, const float4& c) {
            const float dx = q - kvv;
            return c.x + dx * (c.y + dx * (c.z + dx * c.w));
        };
        const float S0 = spl(F0, kv0, c0);
        const float S1 = spl(F1, kv1, c1);
        const float S2 = spl(F2, kv2, c2);
        const float S3 = spl(F3, kv3, c3);

        const float t0v = (float)(j0 + 0) * dt;
        const float t1v = (float)(j0 + 1) * dt;
        const float t2v = (float)(j0 + 2) * dt;
        const float t3v = (float)(j0 + 3) * dt;
        const float w0 = (j0 == 0) ? 0.5f : 1.0f;
        const float w3 = (j0 + 3 == kNT - 1) ? 0.5f : 1.0f;
        const float r0 = t0v - S0, r1 = t1v - S1, r2 = t2v - S2, r3 = t3v - S3;
        acc += w0 * r0 * r0 * fv.x;
        acc +=      r1 * r1 * fv.y;
        acc +=      r2 * r2 * fv.z;
        acc += w3 * r3 * r3 * fv.w;

        // Prefetch tile k+2 into the buffer we just finished reading
        // (all tile reads happened before barrier (B); spline only reads tables).
        if (k + 2 < NTILES) {
            const unsigned lbyte = (unsigned)(tid * 16);
            async_tile_b128(ftr, (unsigned)((k + 2) * TILE * 4) + lbyte,
                            lds_addr_of(&s_tile[cbuf][0]) + lbyte);
        }
    }

    // ---- block reduction of acc -> out[b] ----------------------------------
    #pragma unroll
    for (int off = 16; off >= 1; off >>= 1) acc += __shfl_down(acc, off, 32);
    __syncthreads();
    if (lane == 0) s_red[wv] = acc;
    __syncthreads();
    if (tid == 0) {
        float sum = 0.0f;
        #pragma unroll
        for (int w = 0; w < NWAVES; ++w) sum += s_red[w];
        out[b] = sum * dt;
    }
}

extern "C" void kernel_launch(void* const* d_in, const int* in_sizes, int n_in,
                              void* d_out, int out_size, void* d_ws, size_t ws_size,
                              hipStream_t stream) {
    (void)n_in; (void)d_ws; (void)ws_size; (void)out_size;
    const float* f      = (const float*)d_in[0];   // [B, NT]
    const float* t      = (const float*)d_in[1];   // [NT]
    const float* knots  = (const float*)d_in[2];   // [B, K]
    const float* coeffs = (const float*)d_in[3];   // [B, K-1, 4]
    float* out          = (float*)d_out;           // [B]

    const int B = in_sizes[0] / kNT;               // 2048 for the reference
    w2loss_kernel<<<dim3(B > 0 ? B : 1), dim3(THREADS), 0, stream>>>(
        f, t, knots, coeffs, out);
}